// ClassificationLoss_83726092468740
// MI455X (gfx1250) — compile-verified
//
#include <hip/hip_runtime.h>
#include <hip/hip_bf16.h>

typedef __attribute__((ext_vector_type(2))) float v2f;
typedef __attribute__((ext_vector_type(8))) float v8f;

#define THREADS      256
#define BCE_BLOCKS   512
#define REG_BLOCKS   256
#define HIDDEN_DIM   768
#define PENALTY_HALF (0.5e-4)   // PENALTY * 0.5

// ---------------------------------------------------------------------------
// Kernel 1: BCE partial sums.  softplus(x) - t*x, numerically stable.
// Memory-bound: 163.8 MB stream -> float4 loads, grid-stride, tree reduce.
// ---------------------------------------------------------------------------
__global__ void bce_partial_kernel(const float* __restrict__ logits,
                                   const float* __restrict__ targets,
                                   float* __restrict__ partials,
                                   int n4) {
  const int tid    = blockIdx.x * blockDim.x + threadIdx.x;
  const int stride = gridDim.x * blockDim.x;
  const float4* lg = (const float4*)logits;
  const float4* tg = (const float4*)targets;

  float acc = 0.0f;
  for (int i = tid; i < n4; i += stride) {
    // Prefetch a few strides ahead (speculative; OOB is dropped by HW).
    __builtin_prefetch(&lg[i + 4 * stride], 0, 0);
    __builtin_prefetch(&tg[i + 4 * stride], 0, 0);
    float4 x = lg[i];
    float4 t = tg[i];
    // softplus(x) = max(x,0) + log1p(exp(-|x|))
    acc += fmaxf(x.x, 0.0f) + log1pf(expf(-fabsf(x.x))) - t.x * x.x;
    acc += fmaxf(x.y, 0.0f) + log1pf(expf(-fabsf(x.y))) - t.y * x.y;
    acc += fmaxf(x.z, 0.0f) + log1pf(expf(-fabsf(x.z))) - t.z * x.z;
    acc += fmaxf(x.w, 0.0f) + log1pf(expf(-fabsf(x.w))) - t.w * x.w;
  }

  // wave32 reduce
  for (int off = 16; off > 0; off >>= 1) acc += __shfl_xor(acc, off, 32);

  __shared__ float wsum[THREADS / 32];
  const int lane = threadIdx.x & 31;
  const int wave = threadIdx.x >> 5;
  if (lane == 0) wsum[wave] = acc;
  __syncthreads();
  if (wave == 0) {
    float v = (lane < (THREADS / 32)) ? wsum[lane] : 0.0f;
    for (int off = 16; off > 0; off >>= 1) v += __shfl_xor(v, off, 32);
    if (lane == 0) partials[blockIdx.x] = v;
  }
}

// ---------------------------------------------------------------------------
// Kernel 2: recursive-regularization partial sums via WMMA reduction.
// One wave per edge (scalarized grid-stride: the edge loop runs on SGPRs so
// EXEC provably stays all-1s across every WMMA, and parent/child indices
// become scalar loads).  Each iteration: lane loads float4 of parent & child
// row (coalesced 512B/wave; params is L2-resident: 30.7MB << 192MB L2),
// squares the diffs (v_pk math), and feeds two v_wmma_f32_16x16x4_f32 with
// A = all-ones so C accumulates the tile sums on the matrix pipe.
// ---------------------------------------------------------------------------
__global__ void reg_partial_kernel(const float* __restrict__ params,
                                   const int* __restrict__ parent_idx,
                                   const int* __restrict__ child_idx,
                                   float* __restrict__ partials,
                                   int n_edges) {
  const int lane = threadIdx.x & 31;
  const int wave = threadIdx.x >> 5;
  const int wpb  = blockDim.x >> 5;                  // waves per block

  // Wave-uniform scalar loop bounds: guarantees a scalar (s_cbranch) loop,
  // so EXEC is all-1s for every WMMA, and index loads scalarize.
  const int gwave = __builtin_amdgcn_readfirstlane(blockIdx.x * wpb + wave);
  const int nwave = __builtin_amdgcn_readfirstlane(gridDim.x * wpb);

  v2f ones; ones.x = 1.0f; ones.y = 1.0f;            // A = 16x4 all-ones
  v8f c = {};                                        // persistent accumulator

  for (int e = gwave; e < n_edges; e += nwave) {
    const int pi = __builtin_amdgcn_readfirstlane(parent_idx[e]);
    const int ci = __builtin_amdgcn_readfirstlane(child_idx[e]);
    const float* __restrict__ pr = params + (long)pi * HIDDEN_DIM;
    const float* __restrict__ cr = params + (long)ci * HIDDEN_DIM;
    #pragma unroll
    for (int base = 0; base < HIDDEN_DIM; base += 128) {   // 768/128 = 6 iters
      const int idx = base + lane * 4;
      float4 p = *(const float4*)(pr + idx);
      float4 q = *(const float4*)(cr + idx);
      float dx = p.x - q.x, dy = p.y - q.y;
      float dz = p.z - q.z, dw = p.w - q.w;
      v2f b0; b0.x = dx * dx; b0.y = dy * dy;
      v2f b1; b1.x = dz * dz; b1.y = dw * dw;
      // D = ones(16x4) * B(4x16) + C : every row of D = column sums of B.
      c = __builtin_amdgcn_wmma_f32_16x16x4_f32(false, ones, false, b0,
                                                (short)0, c, false, false);
      c = __builtin_amdgcn_wmma_f32_16x16x4_f32(false, ones, false, b1,
                                                (short)0, c, false, false);
    }
  }

  // c[0] holds D rows M=0 (lanes 0-15) and M=8 (lanes 16-31); rows are
  // identical copies of the column sums, so lane-sum(c[0]) = 2 * total.
  float v = c[0];
  for (int off = 16; off > 0; off >>= 1) v += __shfl_xor(v, off, 32);
  v *= 0.5f;

  __shared__ float wsum[THREADS / 32];
  if (lane == 0) wsum[wave] = v;
  __syncthreads();
  if (wave == 0) {
    float s = (lane < (THREADS / 32)) ? wsum[lane] : 0.0f;
    for (int off = 16; off > 0; off >>= 1) s += __shfl_xor(s, off, 32);
    if (lane == 0) partials[blockIdx.x] = s;
  }
}

// ---------------------------------------------------------------------------
// Kernel 3: final combine (single block, double accumulation for safety).
// out = bce_sum / N  +  0.5 * PENALTY * reg_sum
// ---------------------------------------------------------------------------
__global__ void final_combine_kernel(const float* __restrict__ partials,
                                     int n_bce, int n_reg,
                                     float* __restrict__ out,
                                     double inv_n) {
  double a = 0.0, b = 0.0;
  for (int i = threadIdx.x; i < n_bce; i += blockDim.x) a += (double)partials[i];
  for (int i = threadIdx.x; i < n_reg; i += blockDim.x) b += (double)partials[n_bce + i];

  __shared__ double sa[THREADS];
  __shared__ double sb[THREADS];
  sa[threadIdx.x] = a;
  sb[threadIdx.x] = b;
  __syncthreads();
  for (int s = blockDim.x >> 1; s > 0; s >>= 1) {
    if ((int)threadIdx.x < s) {
      sa[threadIdx.x] += sa[threadIdx.x + s];
      sb[threadIdx.x] += sb[threadIdx.x + s];
    }
    __syncthreads();
  }
  if (threadIdx.x == 0)
    out[0] = (float)(sa[0] * inv_n + PENALTY_HALF * sb[0]);
}

// ---------------------------------------------------------------------------
extern "C" void kernel_launch(void* const* d_in, const int* in_sizes, int n_in,
                              void* d_out, int out_size, void* d_ws, size_t ws_size,
                              hipStream_t stream) {
  const float* logits  = (const float*)d_in[0];
  const float* targets = (const float*)d_in[1];
  const float* params  = (const float*)d_in[2];
  const int*   parent  = (const int*)d_in[3];
  const int*   child   = (const int*)d_in[4];

  const int n_logits = in_sizes[0];          // BATCH * N_LABELS (divisible by 4)
  const int n_edges  = in_sizes[3];          // N_EDGES

  float* ws   = (float*)d_ws;
  float* bceP = ws;                          // [BCE_BLOCKS]
  float* regP = ws + BCE_BLOCKS;             // [REG_BLOCKS]

  bce_partial_kernel<<<BCE_BLOCKS, THREADS, 0, stream>>>(
      logits, targets, bceP, n_logits / 4);

  reg_partial_kernel<<<REG_BLOCKS, THREADS, 0, stream>>>(
      params, parent, child, regP, n_edges);

  final_combine_kernel<<<1, THREADS, 0, stream>>>(
      ws, BCE_BLOCKS, REG_BLOCKS, (float*)d_out, 1.0 / (double)n_logits);
}